// DetModel_16166256902415
// MI455X (gfx1250) — compile-verified
//
#include <hip/hip_runtime.h>
#include <hip/hip_bf16.h>

typedef __attribute__((ext_vector_type(2))) float v2f;
typedef __attribute__((ext_vector_type(8))) float v8f;

#define NB    8192      // batch
#define MDIM  128       // m
#define NDIM  64        // n = 2*TX
#define HID   256
#define PROJ  8

// ---------------------------------------------------------------------------
// Kernel 0: repack weights K-paired so each WMMA B-fragment is ONE b64 load:
//   w?P[k/2][n] = { w[n][k], w[n][k+1] }
// ---------------------------------------------------------------------------
__global__ __launch_bounds__(256) void kTranspose(
    const float* __restrict__ w_ih, const float* __restrict__ w_hh,
    const float* __restrict__ w_x,
    v2f* __restrict__ wihP, v2f* __restrict__ whhP, v2f* __restrict__ wxP) {
  int idx = blockIdx.x * 256 + threadIdx.x;        // 0 .. 32767
  if (idx < 32 * HID) {                            // wih: 64x? -> 32 k-pairs x 256
    int k2 = idx >> 8, n = idx & 255;
    v2f v; v.x = w_ih[n * NDIM + 2 * k2]; v.y = w_ih[n * NDIM + 2 * k2 + 1];
    wihP[idx] = v;
  }
  if (idx < 128 * HID) {                           // whh: 128 k-pairs x 256
    int k2 = idx >> 8, n = idx & 255;
    v2f v; v.x = w_hh[n * HID + 2 * k2]; v.y = w_hh[n * HID + 2 * k2 + 1];
    whhP[idx] = v;
  }
  if (idx < 128 * NDIM) {                          // wx: 128 k-pairs x 64
    int k2 = idx >> 6, i = idx & 63;
    v2f v; v.x = w_x[i * HID + 2 * k2]; v.y = w_x[i * HID + 2 * k2 + 1];
    wxP[idx] = v;
  }
}

// ---------------------------------------------------------------------------
// Kernel 1: per-batch HtH (64x64) via f32 WMMA, Hty, and state init.
// H staged TRANSPOSED in LDS: sHT[n][k], stride 132 (mod 64 = 4) so each
// fragment is one aligned ds_load_b64 and half-waves (k, k+2) hit disjoint
// bank pairs. 1 workgroup (128 thr = 4 waves) per batch element.
// ---------------------------------------------------------------------------
#define LDHT 132
__global__ __launch_bounds__(128) void kPrecompute(
    const float* __restrict__ y, const float* __restrict__ H,
    const float* __restrict__ x0, const float* __restrict__ h0,
    float* __restrict__ HtH, float* __restrict__ Hty,
    float* __restrict__ xbuf, float* __restrict__ out_h) {
  __shared__ float sHT[NDIM * LDHT];   // 64 x 132 = 33.8 KB
  __shared__ float sy[MDIM];

  const int tid = threadIdx.x;
  const size_t b = blockIdx.x;
  const float* Hb = H + b * (size_t)(MDIM * NDIM);

  // Stage H[b] (128x64 row-major) transposed into LDS.
  const float4* H4 = (const float4*)Hb;
  for (int idx = tid; idx < (MDIM * NDIM) / 4; idx += 128) {
    float4 v = H4[idx];
    int r = idx >> 4, c = (idx & 15) << 2;         // H[r][c..c+3]
    sHT[(c + 0) * LDHT + r] = v.x;
    sHT[(c + 1) * LDHT + r] = v.y;
    sHT[(c + 2) * LDHT + r] = v.z;
    sHT[(c + 3) * LDHT + r] = v.w;
  }
  sy[tid] = y[b * MDIM + tid];
  __syncthreads();

  const int wave = tid >> 5, lane = tid & 31;
  const int l = lane & 15;
  const int kh = (lane >> 4) << 1;      // half-wave K offset: 0 or 2
  const int mt = wave;                  // wave owns 16-row strip of output

  v8f acc[4] = {};
  for (int k0 = 0; k0 < MDIM; k0 += 4) {
    // A[m][k] = H[k][m] = sHT[m][k]  -> one b64 load
    v2f a = *(const v2f*)&sHT[(mt * 16 + l) * LDHT + k0 + kh];
#pragma unroll
    for (int nt = 0; nt < 4; ++nt) {
      // B[k][j] = H[k][j] = sHT[j][k] -> one b64 load
      v2f bf = *(const v2f*)&sHT[(nt * 16 + l) * LDHT + k0 + kh];
      acc[nt] = __builtin_amdgcn_wmma_f32_16x16x4_f32(
          false, a, false, bf, (short)0, acc[nt], false, false);
    }
  }

  float* out = HtH + b * (size_t)(NDIM * NDIM);
#pragma unroll
  for (int nt = 0; nt < 4; ++nt)
#pragma unroll
    for (int r = 0; r < 8; ++r) {
      int i = mt * 16 + r + ((lane >> 4) << 3);
      int j = nt * 16 + l;
      out[i * NDIM + j] = acc[nt][r];
    }

  // Hty + x init (threads 0..63), h init (all 128 threads, 2 elems each)
  if (tid < NDIM) {
    float s = 0.f;
    for (int m = 0; m < MDIM; ++m) s = fmaf(sHT[tid * LDHT + m], sy[m], s);
    Hty[b * NDIM + tid]  = s;
    xbuf[b * NDIM + tid] = x0[b * NDIM + tid];
  }
  out_h[b * HID + tid]       = h0[b * HID + tid];
  out_h[b * HID + 128 + tid] = h0[b * HID + 128 + tid];
}

// ---------------------------------------------------------------------------
// Kernel 2: 10 gradient-descent steps  x += 2s*(Hty - HtH x).
// Mat*vec (N=1) -> VALU FMAs out of LDS. 2 batches per 128-thr workgroup.
// ---------------------------------------------------------------------------
#define LDA 65   // HtH LDS stride: bank = (t + k) % 64, conflict-free
__global__ __launch_bounds__(128) void kGD(
    const float* __restrict__ HtH, const float* __restrict__ Hty,
    float* __restrict__ xbuf, const float* __restrict__ step,
    const int* __restrict__ itersPtr) {
  __shared__ float sA[2 * NDIM * LDA];
  __shared__ float sx[2 * NDIM];

  const int tid = threadIdx.x;
  const int sub = tid >> 6, t = tid & 63;
  const size_t b = (size_t)blockIdx.x * 2 + sub;
  const float* Ab = HtH + b * (size_t)(NDIM * NDIM);
  float* sAs = sA + sub * NDIM * LDA;

  for (int idx = t; idx < NDIM * NDIM; idx += 64)
    sAs[(idx >> 6) * LDA + (idx & 63)] = Ab[idx];

  const float hty   = Hty[b * NDIM + t];
  const float two_s = 2.0f * step[0];
  const int   iters = itersPtr[0];
  sx[sub * NDIM + t] = xbuf[b * NDIM + t];
  __syncthreads();

  for (int it = 0; it < iters; ++it) {
    float dot = 0.f;
#pragma unroll 16
    for (int k = 0; k < NDIM; ++k)
      dot = fmaf(sAs[t * LDA + k], sx[sub * NDIM + k], dot);
    float xn = sx[sub * NDIM + t] + two_s * (hty - dot);
    __syncthreads();
    sx[sub * NDIM + t] = xn;
    __syncthreads();
  }
  xbuf[b * NDIM + t] = sx[sub * NDIM + t];
}

// ---------------------------------------------------------------------------
// Kernel 3: RNN cell + projection, 16 batch rows per 256-thr workgroup.
//   h' = lrelu(z wihT + h whhT + b_ih + b_hh);  xf = h' wxT + b_x
// A-fragments: single b64 LDS loads (k,k+1 contiguous, even k, strides keep
// 8B alignment). B-fragments: single b64 global loads from K-paired weights.
// ---------------------------------------------------------------------------
#define LDZ  68   // 64+4: half-wave A reads rotate 4 banks/row
#define LDHS 260  // 256+4
__global__ __launch_bounds__(256) void kRNN(
    float* __restrict__ xbuf, float* __restrict__ hio,
    const v2f* __restrict__ wihP, const v2f* __restrict__ whhP,
    const v2f* __restrict__ wxP,
    const float* __restrict__ b_ih, const float* __restrict__ b_hh,
    const float* __restrict__ b_x,
    float* __restrict__ out_seq, float* __restrict__ out_xf, int p) {
  __shared__ float sz[16 * LDZ];
  __shared__ float sh[16 * LDHS];

  const int tid = threadIdx.x;
  const int b0 = blockIdx.x * 16;

  for (int idx = tid; idx < 16 * NDIM; idx += 256) {
    int r = idx >> 6, c = idx & 63;
    sz[r * LDZ + c] = xbuf[(size_t)(b0 + r) * NDIM + c];
  }
  for (int idx = tid; idx < 16 * HID; idx += 256) {
    int r = idx >> 8, c = idx & 255;
    sh[r * LDHS + c] = hio[(size_t)(b0 + r) * HID + c];
  }
  __syncthreads();

  const int wave = tid >> 5, lane = tid & 31;
  const int l = lane & 15;
  const int kh = (lane >> 4) << 1;          // 0 or 2
  const int k2h = lane >> 4;                // k-pair offset: 0 or 1
  const int n0 = (wave * 2 + 0) * 16 + l;   // hidden col, tile 0
  const int n1 = (wave * 2 + 1) * 16 + l;   // hidden col, tile 1

  v8f acc0 = {}, acc1 = {};
  // z @ wihT  (K = 64)
  for (int k0 = 0; k0 < NDIM; k0 += 4) {
    v2f a = *(const v2f*)&sz[l * LDZ + k0 + kh];
    const int k2 = (k0 >> 1) + k2h;
    v2f bA = wihP[k2 * HID + n0];
    v2f bB = wihP[k2 * HID + n1];
    acc0 = __builtin_amdgcn_wmma_f32_16x16x4_f32(false, a, false, bA, (short)0, acc0, false, false);
    acc1 = __builtin_amdgcn_wmma_f32_16x16x4_f32(false, a, false, bB, (short)0, acc1, false, false);
  }
  // h @ whhT  (K = 256)
  for (int k0 = 0; k0 < HID; k0 += 4) {
    v2f a = *(const v2f*)&sh[l * LDHS + k0 + kh];
    const int k2 = (k0 >> 1) + k2h;
    v2f bA = whhP[k2 * HID + n0];
    v2f bB = whhP[k2 * HID + n1];
    acc0 = __builtin_amdgcn_wmma_f32_16x16x4_f32(false, a, false, bA, (short)0, acc0, false, false);
    acc1 = __builtin_amdgcn_wmma_f32_16x16x4_f32(false, a, false, bB, (short)0, acc1, false, false);
  }

  const float bias0 = b_ih[n0] + b_hh[n0];
  const float bias1 = b_ih[n1] + b_hh[n1];

  __syncthreads();   // everyone done reading old h before overwrite
#pragma unroll
  for (int r = 0; r < 8; ++r) {
    int m = r + ((lane >> 4) << 3);
    float v0 = acc0[r] + bias0; v0 = v0 > 0.f ? v0 : 0.01f * v0;
    float v1 = acc1[r] + bias1; v1 = v1 > 0.f ? v1 : 0.01f * v1;
    sh[m * LDHS + n0] = v0;
    sh[m * LDHS + n1] = v1;
    hio[(size_t)(b0 + m) * HID + n0] = v0;
    hio[(size_t)(b0 + m) * HID + n1] = v1;
  }
  __syncthreads();

  // xf = h' @ wxT  (K = 256, 4 output tiles -> waves 0..3; wave-uniform branch)
  if (wave < 4) {
    v8f acc = {};
    const int j = wave * 16 + l;
    for (int k0 = 0; k0 < HID; k0 += 4) {
      v2f a = *(const v2f*)&sh[l * LDHS + k0 + kh];
      const int k2 = (k0 >> 1) + k2h;
      v2f bf = wxP[k2 * NDIM + j];
      acc = __builtin_amdgcn_wmma_f32_16x16x4_f32(false, a, false, bf, (short)0, acc, false, false);
    }
    const float bb = b_x[j];
#pragma unroll
    for (int r = 0; r < 8; ++r) {
      int m = r + ((lane >> 4) << 3);
      float v = acc[r] + bb;
      out_seq[(size_t)p * NB * NDIM + (size_t)(b0 + m) * NDIM + j] = v;
      xbuf[(size_t)(b0 + m) * NDIM + j] = v;
      if (p == PROJ - 1) out_xf[(size_t)(b0 + m) * NDIM + j] = v;
    }
  }
}

// ---------------------------------------------------------------------------
extern "C" void kernel_launch(void* const* d_in, const int* in_sizes, int n_in,
                              void* d_out, int out_size, void* d_ws, size_t ws_size,
                              hipStream_t stream) {
  const float* y     = (const float*)d_in[0];
  const float* H     = (const float*)d_in[1];
  const float* x0    = (const float*)d_in[2];
  const float* h0    = (const float*)d_in[3];
  const float* step  = (const float*)d_in[4];
  const float* w_ih  = (const float*)d_in[5];
  const float* w_hh  = (const float*)d_in[6];
  const float* b_ih  = (const float*)d_in[7];
  const float* b_hh  = (const float*)d_in[8];
  const float* w_x   = (const float*)d_in[9];
  const float* b_x   = (const float*)d_in[10];
  const int*   iters = (const int*)d_in[11];

  float* ws   = (float*)d_ws;
  float* HtH  = ws;                                    // NB*64*64
  float* Hty  = HtH + (size_t)NB * NDIM * NDIM;        // NB*64
  float* xbuf = Hty + (size_t)NB * NDIM;               // NB*64
  float* wtop = xbuf + (size_t)NB * NDIM;
  v2f* wihP = (v2f*)wtop;                              // 32*256 v2f
  v2f* whhP = wihP + 32 * HID;                         // 128*256 v2f
  v2f* wxP  = whhP + 128 * HID;                        // 128*64 v2f

  float* out_xf  = (float*)d_out;                      // NB*64
  float* out_h   = out_xf + (size_t)NB * NDIM;         // NB*256
  float* out_seq = out_h + (size_t)NB * HID;           // PROJ*NB*64

  kTranspose<<<128, 256, 0, stream>>>(w_ih, w_hh, w_x, wihP, whhP, wxP);
  kPrecompute<<<NB, 128, 0, stream>>>(y, H, x0, h0, HtH, Hty, xbuf, out_h);
  for (int p = 0; p < PROJ; ++p) {
    kGD<<<NB / 2, 128, 0, stream>>>(HtH, Hty, xbuf, step, iters);
    kRNN<<<NB / 16, 256, 0, stream>>>(xbuf, out_h, wihP, whhP, wxP,
                                      b_ih, b_hh, b_x, out_seq, out_xf, p);
  }
}